// TemporalAttention_8753143349702
// MI455X (gfx1250) — compile-verified
//
#include <hip/hip_runtime.h>
#include <math.h>

// Problem constants (from reference): b=16, T=32, C = 128*32*32 = 131072.
#define BATCH   16
#define TSTEPS  32
#define CFLAT   131072
#define NSLICE  8
#define SLICE   (CFLAT / NSLICE)      // 16384 floats per score slice
#define KBLKS   (CFLAT / 128)         // 1024 k-blocks of 128 per batch (8 waves x 16)

typedef __attribute__((ext_vector_type(4))) float fvec4;
typedef __attribute__((ext_vector_type(2))) float v2f;
typedef __attribute__((ext_vector_type(8))) float v8f;

// ---------------------------------------------------------------------------
// Pass 1: partial scores.  partial[(b*T + t)*NSLICE + s] =
//         sum over slice s of curr[b,k] * hist[t,b,k].
// History is streamed non-temporally (single use); current stays cacheable
// (re-read 32x, 8 MB fits easily in the 192 MB L2).
// ---------------------------------------------------------------------------
__global__ __launch_bounds__(256)
void ta_scores_kernel(const float* __restrict__ curr,
                      const float* __restrict__ hist,
                      float* __restrict__ partial) {
  const int blk   = blockIdx.x;          // = (b*32 + t)*8 + slice
  const int slice = blk & (NSLICE - 1);
  const int t     = (blk >> 3) & (TSTEPS - 1);
  const int b     = blk >> 8;
  const int tid   = threadIdx.x;

  const fvec4* c4 = (const fvec4*)(curr + (size_t)b * CFLAT + (size_t)slice * SLICE);
  const fvec4* h4 = (const fvec4*)(hist + ((size_t)t * BATCH + b) * CFLAT
                                        + (size_t)slice * SLICE);

  float acc = 0.0f;
#pragma unroll
  for (int i = 0; i < SLICE / 4 / 256; ++i) {          // 16 float4 per thread
    fvec4 h = __builtin_nontemporal_load(&h4[tid + i * 256]);
    fvec4 c = c4[tid + i * 256];
    acc += h.x * c.x + h.y * c.y + h.z * c.z + h.w * c.w;
  }

  // wave32 shuffle reduction, then cross-wave via LDS
  for (int off = 16; off > 0; off >>= 1)
    acc += __shfl_xor(acc, off, 32);

  __shared__ float wsum[8];
  if ((tid & 31) == 0) wsum[tid >> 5] = acc;
  __syncthreads();
  if (tid == 0) {
    float s = 0.0f;
#pragma unroll
    for (int w = 0; w < 8; ++w) s += wsum[w];
    partial[blk] = s;
  }
}

// ---------------------------------------------------------------------------
// Pass 2: per-batch softmax over T=32 (one wave32 per batch element).
// ---------------------------------------------------------------------------
__global__ __launch_bounds__(32)
void ta_softmax_kernel(const float* __restrict__ partial,
                       float* __restrict__ weights) {
  const int b = blockIdx.x;
  const int t = threadIdx.x;             // 0..31

  float s = 0.0f;
#pragma unroll
  for (int i = 0; i < NSLICE; ++i)
    s += partial[(b * TSTEPS + t) * NSLICE + i];
  s *= 0.2f;                             // / SCALE (5.0)

  float m = s;
  for (int off = 16; off > 0; off >>= 1)
    m = fmaxf(m, __shfl_xor(m, off, 32));
  float e = expf(s - m);
  float sum = e;
  for (int off = 16; off > 0; off >>= 1)
    sum += __shfl_xor(sum, off, 32);
  weights[b * TSTEPS + t] = e / sum;
}

// ---------------------------------------------------------------------------
// Pass 3: weighted history sum via v_wmma_f32_16x16x4_f32.
//   out[b, k0+n] = sum_t w[b,t] * hist[t,b,k0+n]
// A(16x4) = weights broadcast down rows, B(4x16) = history tile, so every row
// of D holds the 16 finished outputs; accumulate over 8 tc-chunks (T=32).
// A layout (ISA 7.12.2, 32-bit A 16x4): lanes 0-15 hold K=0,1; lanes 16-31
// hold K=2,3.  B rows follow the same half-wave K split, one row of 16
// consecutive k per 16 lanes -> 64B contiguous segments, tiled into full
// lines by the 8 waves of the block covering 128 consecutive k.
// ---------------------------------------------------------------------------
__global__ __launch_bounds__(256)
void ta_weighted_sum_kernel(const float* __restrict__ hist,
                            const float* __restrict__ weights,
                            float* __restrict__ out) {
  const int blk  = blockIdx.x;           // = b*KBLKS + kb
  const int kb   = blk & (KBLKS - 1);
  const int b    = blk >> 10;
  const int wave = threadIdx.x >> 5;
  const int lane = threadIdx.x & 31;
  const int hi   = lane >> 4;            // which half-wave (K pair 0/1 vs 2/3)
  const int n    = lane & 15;            // output column within the 16-block
  const int k0   = kb * 128 + wave * 16;

  const float*  wb = weights + b * TSTEPS;
  const size_t  bC = (size_t)b * CFLAT;

  v8f acc = {};                          // C/D accumulator = 0
#pragma unroll
  for (int tc = 0; tc < TSTEPS; tc += 4) {
    const int t0 = tc + 2 * hi;

    v2f a;                               // A: weights, identical for all rows
    a.x = wb[t0];
    a.y = wb[t0 + 1];

    v2f bf;                              // B: hist rows t0, t0+1 at column k0+n
    bf.x = __builtin_nontemporal_load(hist + ((size_t)t0       * BATCH) * CFLAT + bC + k0 + n);
    bf.y = __builtin_nontemporal_load(hist + ((size_t)(t0 + 1) * BATCH) * CFLAT + bC + k0 + n);

    acc = __builtin_amdgcn_wmma_f32_16x16x4_f32(
        /*neg_a=*/false, a, /*neg_b=*/false, bf,
        /*c_mod=*/(short)0, acc, /*reuse_a=*/false, /*reuse_b=*/false);
  }

  // Every row of D is identical; VGPR0 at lane L holds column (L & 15).
  if (hi == 0)
    out[bC + k0 + n] = acc[0];
}

// ---------------------------------------------------------------------------
extern "C" void kernel_launch(void* const* d_in, const int* in_sizes, int n_in,
                              void* d_out, int out_size, void* d_ws, size_t ws_size,
                              hipStream_t stream) {
  const float* curr = (const float*)d_in[0];   // [16,128,32,32]
  const float* hist = (const float*)d_in[1];   // [32,16,128,32,32]
  float*       out  = (float*)d_out;           // [16,128,32,32]

  float* partial = (float*)d_ws;                       // 16*32*8 floats
  float* weights = partial + BATCH * TSTEPS * NSLICE;  // 16*32 floats

  ta_scores_kernel<<<BATCH * TSTEPS * NSLICE, 256, 0, stream>>>(curr, hist, partial);
  ta_softmax_kernel<<<BATCH, 32, 0, stream>>>(partial, weights);
  ta_weighted_sum_kernel<<<BATCH * KBLKS, 256, 0, stream>>>(hist, weights, out);
}